// SWAStatefulChunk1_9302899163266
// MI455X (gfx1250) — compile-verified
//
#include <hip/hip_runtime.h>
#include <math.h>

// Problem constants (must match reference)
#define DMODEL 2048
#define NH 8
#define NKV 2
#define HD 256
#define DFF 8192
#define WCTX 512
#define PLD 256
#define EPSF 1e-6f

typedef __attribute__((ext_vector_type(2))) float v2f;
typedef __attribute__((ext_vector_type(8))) float v8f;

// ---------------- reductions (wave32) ----------------
__device__ __forceinline__ float warp_sum(float v) {
#pragma unroll
  for (int o = 16; o > 0; o >>= 1) v += __shfl_down(v, o, 32);
  return v;
}

__device__ float block_sum(float v, float* sm) {
  const int lane = threadIdx.x & 31;
  const int wv = threadIdx.x >> 5;
  v = warp_sum(v);
  if (lane == 0) sm[wv] = v;
  __syncthreads();
  const int nw = (blockDim.x + 31) >> 5;
  if (wv == 0) {
    float t = (lane < nw) ? sm[lane] : 0.0f;
    t = warp_sum(t);
    if (lane == 0) sm[0] = t;
  }
  __syncthreads();
  float r = sm[0];
  __syncthreads();   // allow sm reuse by caller
  return r;
}

__device__ __forceinline__ float gelu_tanh(float x) {
  const float c0 = 0.7978845608028654f;  // sqrt(2/pi)
  float x3 = x * x * x;
  return 0.5f * x * (1.0f + tanhf(c0 * (x + 0.044715f * x3)));
}

// ---------------- RMSNorm (+optional weight/residual/scalar) ----------------
__global__ void rmsnorm_kernel(const float* __restrict__ in,
                               const float* __restrict__ w,      // may be null
                               const float* __restrict__ resid,  // may be null
                               const float* __restrict__ scale,  // may be null (1 elem)
                               float* __restrict__ out, int n) {
  __shared__ float sm[32];
  float ss = 0.0f;
  for (int i = threadIdx.x; i < n; i += blockDim.x) {
    float v = in[i];
    ss += v * v;
  }
  ss = block_sum(ss, sm);
  float r = rsqrtf(ss / (float)n + EPSF);
  float sc = scale ? scale[0] : 1.0f;
  for (int i = threadIdx.x; i < n; i += blockDim.x) {
    float v = in[i] * r;
    if (w) v *= (1.0f + w[i]);
    if (resid) v += resid[i];
    out[i] = v * sc;
  }
}

// ---------------- streaming GEMV: out[row] = dot(W[row,:], x) ----------------
__global__ void gemv_kernel(const float* __restrict__ Wm, const float* __restrict__ x,
                            float* __restrict__ out, int n) {
  __shared__ float sm[32];
  const int row = blockIdx.x;
  const float4* __restrict__ w4 = (const float4*)(Wm + (size_t)row * n);
  const float4* __restrict__ x4 = (const float4*)x;
  const int nf4 = n >> 2;
  float acc = 0.0f;
  for (int i = threadIdx.x; i < nf4; i += blockDim.x) {
    __builtin_prefetch(&w4[i + 2 * blockDim.x], 0, 0);  // global_prefetch_b8, NT stream
    float4 a = w4[i];
    float4 b = x4[i];
    acc += a.x * b.x + a.y * b.y + a.z * b.z + a.w * b.w;
  }
  acc = block_sum(acc, sm);
  if (threadIdx.x == 0) out[row] = acc;
}

// gate/up fused: act[row] = gelu(dot(Wg,x)) * dot(Wu,x)
__global__ void gateup_kernel(const float* __restrict__ Wg, const float* __restrict__ Wu,
                              const float* __restrict__ x, float* __restrict__ act, int n) {
  __shared__ float sm[32];
  const int row = blockIdx.x;
  const float4* __restrict__ g4 = (const float4*)(Wg + (size_t)row * n);
  const float4* __restrict__ u4 = (const float4*)(Wu + (size_t)row * n);
  const float4* __restrict__ x4 = (const float4*)x;
  const int nf4 = n >> 2;
  float ag = 0.0f, au = 0.0f;
  for (int i = threadIdx.x; i < nf4; i += blockDim.x) {
    __builtin_prefetch(&g4[i + 2 * blockDim.x], 0, 0);
    __builtin_prefetch(&u4[i + 2 * blockDim.x], 0, 0);
    float4 xv = x4[i];
    float4 g = g4[i];
    float4 u = u4[i];
    ag += g.x * xv.x + g.y * xv.y + g.z * xv.z + g.w * xv.w;
    au += u.x * xv.x + u.y * xv.y + u.z * xv.z + u.w * xv.w;
  }
  ag = block_sum(ag, sm);
  au = block_sum(au, sm);
  if (threadIdx.x == 0) act[row] = gelu_tanh(ag) * au;
}

// pl_gate: out[row] = gelu(dot(W,x)) * mul[row]
__global__ void gemv_gelu_mul_kernel(const float* __restrict__ Wm, const float* __restrict__ x,
                                     const float* __restrict__ mul, float* __restrict__ out, int n) {
  __shared__ float sm[32];
  const int row = blockIdx.x;
  const float4* __restrict__ w4 = (const float4*)(Wm + (size_t)row * n);
  const float4* __restrict__ x4 = (const float4*)x;
  const int nf4 = n >> 2;
  float acc = 0.0f;
  for (int i = threadIdx.x; i < nf4; i += blockDim.x) {
    float4 a = w4[i];
    float4 b = x4[i];
    acc += a.x * b.x + a.y * b.y + a.z * b.z + a.w * b.w;
  }
  acc = block_sum(acc, sm);
  if (threadIdx.x == 0) out[row] = gelu_tanh(acc) * mul[row];
}

// ---------------- per-head RMSNorm + RoPE (q/k), weightless norm (v) ----------------
// grid: 12 blocks x 256 threads. 0-7: q heads, 8-9: k heads, 10-11: v heads.
__global__ void headnorm_rope_kernel(const float* __restrict__ qkv,
                                     const float* __restrict__ qw, const float* __restrict__ kw,
                                     const float* __restrict__ cosv, const float* __restrict__ sinv,
                                     float* __restrict__ qf, float* __restrict__ kf,
                                     float* __restrict__ vf) {
  __shared__ float sm[32];
  __shared__ float sn[HD];
  const int b = blockIdx.x;
  const float* src;
  float* dst;
  const float* w;
  int rope;
  if (b < 8) {
    src = qkv + b * HD; dst = qf + b * HD; w = qw; rope = 1;
  } else if (b < 10) {
    src = qkv + DMODEL + (b - 8) * HD; dst = kf + (b - 8) * HD; w = kw; rope = 1;
  } else {
    src = qkv + DMODEL + NKV * HD + (b - 10) * HD; dst = vf + (b - 10) * HD; w = nullptr; rope = 0;
  }
  const int t = threadIdx.x;  // 256 == HD
  float v = src[t];
  float ss = block_sum(v * v, sm);
  float r = rsqrtf(ss / (float)HD + EPSF);
  float nv = v * r;
  if (w) nv *= (1.0f + w[t]);
  if (rope) {
    sn[t] = nv;
    __syncthreads();
    float rot = (t < 128) ? -sn[t + 128] : sn[t - 128];
    dst[t] = nv * cosv[t] + rot * sinv[t];
  } else {
    dst[t] = nv;
  }
}

// ---------------- GQA attention via V_WMMA_F32_16X16X4_F32 ----------------
// One block per KV group (grid=2), 512 threads = 16 wave32s.
// A matrix is padded to 16 rows (rows 4..15 zero) so the WMMA inner loops are
// completely branch-free: ds_load_b64 (A) + global_load_b64 (B) + wmma.
// ring_pos handling: pointer-select once per tile in phase B; rank-1
// correction O += P[:,ring] * (v_new - V_cache[ring]) at store in phase D.
#define SQ_STRIDE (HD + 2)
#define SS_STRIDE (WCTX + 2)

__global__ void attention_kernel(const float* __restrict__ qf, const float* __restrict__ kf,
                                 const float* __restrict__ vf, const float* __restrict__ kvc,
                                 const float* __restrict__ mask, const int* __restrict__ ring_p,
                                 float* __restrict__ o) {
  __shared__ float sQ[16][SQ_STRIDE];   // ~16.5 KB, rows 4..15 zero
  __shared__ float sS[16][SS_STRIDE];   // ~33 KB, rows 4..15 written-but-unused
  const int g = blockIdx.x;
  const int tid = threadIdx.x;
  const int wv = tid >> 5;
  const int lane = tid & 31;
  const int row = lane & 15;      // M index (A rows) or N index (B/C cols)
  const int hi = lane >> 4;       // lane half selects K pair {0,1} vs {2,3}
  const int ring = ring_p[0];

  // Zero-padded A: rows 0..3 = this group's 4 query heads, rows 4..15 = 0.
  for (int i = tid; i < 16 * HD; i += blockDim.x) {
    sQ[i >> 8][i & 255] = (i < 4 * HD) ? qf[g * 4 * HD + i] : 0.0f;
  }
  __syncthreads();

  const float* __restrict__ Kc = kvc + (size_t)(0 * NKV + g) * WCTX * HD;
  const float* __restrict__ Vc = kvc + (size_t)(1 * NKV + g) * WCTX * HD;
  const float* __restrict__ kn = kf + g * HD;
  const float* __restrict__ vn = vf + g * HD;

  // ---- Phase B: S = Q * K^T  (branch-free inner loop) ----
  for (int tile = wv; tile < WCTX / 16; tile += 16) {
    const int wbase = tile * 16;
    const int wpos = wbase + row;                         // B column index n
    // select K source once per tile: fresh k for the ring slot, cache otherwise
    const float* __restrict__ src = (wpos == ring) ? kn : (Kc + (size_t)wpos * HD);
    v8f acc = {};
    for (int k4 = 0; k4 < HD; k4 += 4) {
      const int d = k4 + 2 * hi;
      v2f a, bb;
      a.x = sQ[row][d];         // A(M=row, K=d)     -> ds_load_b64
      a.y = sQ[row][d + 1];
      bb.x = src[d];            // B(K=d, N=row)     -> global_load_b64
      bb.y = src[d + 1];
      acc = __builtin_amdgcn_wmma_f32_16x16x4_f32(false, a, false, bb, (short)0, acc,
                                                  false, false);
    }
    const float mk = mask[wpos];
#pragma unroll
    for (int j = 0; j < 8; j++) {
      const int M = j + 8 * hi;                           // C/D row per ISA layout
      sS[M][wbase + row] = acc[j] + mk;                   // unconditional store
    }
  }
  __syncthreads();

  // ---- Phase C: softmax over rows 0..3 (one wave per head) ----
  if (wv < 4) {
    float* __restrict__ srow = sS[wv];
    float mx = -3.0e38f;
    for (int i = lane; i < WCTX; i += 32) mx = fmaxf(mx, srow[i]);
#pragma unroll
    for (int off = 16; off > 0; off >>= 1) mx = fmaxf(mx, __shfl_down(mx, off, 32));
    mx = __shfl(mx, 0, 32);
    float sum = 0.0f;
    for (int i = lane; i < WCTX; i += 32) {
      float e = __expf(srow[i] - mx);
      srow[i] = e;
      sum += e;
    }
    sum = warp_sum(sum);
    sum = __shfl(sum, 0, 32);
    float inv = 1.0f / sum;
    for (int i = lane; i < WCTX; i += 32) srow[i] *= inv;
  }
  __syncthreads();

  // ---- Phase D: O = P * V_cache, then rank-1 ring correction at store ----
  {
    const int dbase = wv * 16;
    const int dd = dbase + row;                           // output column d
    v8f acc = {};
    for (int k4 = 0; k4 < WCTX; k4 += 4) {
      const int w0 = k4 + 2 * hi;
      v2f a, bb;
      a.x = sS[row][w0];                                  // A(M=row, K=w0)
      a.y = sS[row][w0 + 1];
      bb.x = Vc[(size_t)w0 * HD + dd];                    // B(K=w0, N=row), coalesced
      bb.y = Vc[(size_t)(w0 + 1) * HD + dd];
      acc = __builtin_amdgcn_wmma_f32_16x16x4_f32(false, a, false, bb, (short)0, acc,
                                                  false, false);
    }
    // correction term: P[M][ring] * (v_new[dd] - V_cache[ring][dd])
    const float dv = vn[dd] - Vc[(size_t)ring * HD + dd];
#pragma unroll
    for (int j = 0; j < 8; j++) {
      const int M = j + 8 * hi;
      if (M < 4) {
        o[(size_t)(g * 4 + M) * HD + dd] = acc[j] + sS[M][ring] * dv;
      }
    }
  }
}

// ---------------- host-side orchestration ----------------
extern "C" void kernel_launch(void* const* d_in, const int* in_sizes, int n_in,
                              void* d_out, int out_size, void* d_ws, size_t ws_size,
                              hipStream_t stream) {
  (void)in_sizes; (void)n_in; (void)out_size; (void)ws_size;
  const float* hidden  = (const float*)d_in[0];
  const float* cosv    = (const float*)d_in[1];
  const float* sinv    = (const float*)d_in[2];
  const float* mask    = (const float*)d_in[3];
  const float* kvc     = (const float*)d_in[4];
  const float* perl    = (const float*)d_in[5];
  const float* ln_in_w = (const float*)d_in[6];
  const float* Wq      = (const float*)d_in[7];
  const float* q_nw    = (const float*)d_in[8];
  const float* Wk      = (const float*)d_in[9];
  const float* Wv      = (const float*)d_in[10];
  const float* k_nw    = (const float*)d_in[11];
  const float* Wo      = (const float*)d_in[12];
  const float* pa_w    = (const float*)d_in[13];
  const float* pf_w    = (const float*)d_in[14];
  const float* Wgate   = (const float*)d_in[15];
  const float* Wup     = (const float*)d_in[16];
  const float* Wdown   = (const float*)d_in[17];
  const float* pffn_w  = (const float*)d_in[18];
  const float* Wplg    = (const float*)d_in[19];
  const float* Wplp    = (const float*)d_in[20];
  const float* ppl_w   = (const float*)d_in[21];
  const float* lscal   = (const float*)d_in[22];
  const int*   ringp   = (const int*)d_in[23];
  float* out = (float*)d_out;

  float* ws  = (float*)d_ws;
  float* h0       = ws;               // 2048
  float* qkv      = h0 + 2048;        // 3072
  float* qf       = qkv + 3072;       // 2048
  float* kf       = qf + 2048;        // 512
  float* vf       = kf + 512;         // 512
  float* o        = vf + 512;         // 2048
  float* attn_raw = o + 2048;         // 2048
  float* x1       = attn_raw + 2048;  // 2048
  float* hn       = x1 + 2048;        // 2048
  float* act      = hn + 2048;        // 8192
  float* mlp      = act + 8192;       // 2048
  float* x2       = mlp + 2048;       // 2048
  float* plg      = x2 + 2048;        // 256
  float* plp      = plg + 256;        // 2048

  const int TB = 256;

  // 1. input layernorm
  rmsnorm_kernel<<<1, TB, 0, stream>>>(hidden, ln_in_w, nullptr, nullptr, h0, DMODEL);
  // 2. q/k/v projections (bandwidth-bound GEMVs)
  gemv_kernel<<<NH * HD, TB, 0, stream>>>(Wq, h0, qkv, DMODEL);
  gemv_kernel<<<NKV * HD, TB, 0, stream>>>(Wk, h0, qkv + DMODEL, DMODEL);
  gemv_kernel<<<NKV * HD, TB, 0, stream>>>(Wv, h0, qkv + DMODEL + NKV * HD, DMODEL);
  // 3. per-head norms + RoPE
  headnorm_rope_kernel<<<12, TB, 0, stream>>>(qkv, q_nw, k_nw, cosv, sinv, qf, kf, vf);
  // 4. attention (WMMA), ring_pos substitution handled without cache copy
  attention_kernel<<<NKV, 512, 0, stream>>>(qf, kf, vf, kvc, mask, ringp, o);
  // 5. output projection + post-attn norm + residual
  gemv_kernel<<<DMODEL, TB, 0, stream>>>(Wo, o, attn_raw, NH * HD);
  rmsnorm_kernel<<<1, TB, 0, stream>>>(attn_raw, pa_w, hidden, nullptr, x1, DMODEL);
  // 6. MLP
  rmsnorm_kernel<<<1, TB, 0, stream>>>(x1, pf_w, nullptr, nullptr, hn, DMODEL);
  gateup_kernel<<<DFF, TB, 0, stream>>>(Wgate, Wup, hn, act, DMODEL);
  gemv_kernel<<<DMODEL, TB, 0, stream>>>(Wdown, act, mlp, DFF);
  rmsnorm_kernel<<<1, TB, 0, stream>>>(mlp, pffn_w, x1, nullptr, x2, DMODEL);
  // 7. per-layer gate/projection
  gemv_gelu_mul_kernel<<<PLD, TB, 0, stream>>>(Wplg, x2, perl, plg, DMODEL);
  gemv_kernel<<<DMODEL, TB, 0, stream>>>(Wplp, plg, plp, PLD);
  rmsnorm_kernel<<<1, TB, 0, stream>>>(plp, ppl_w, x2, lscal, out, DMODEL);
}